// S2VT_24996709662941
// MI455X (gfx1250) — compile-verified
//
#include <hip/hip_runtime.h>
#include <hip/hip_bf16.h>
#include <math.h>

#define HID   512
#define GATES 2048
#define VOCAB 3003
#define TIN   80
#define TOUT  40
#define INDIM 4096

typedef float v2f __attribute__((ext_vector_type(2)));
typedef float v8f __attribute__((ext_vector_type(8)));

// ---------------------------------------------------------------------------
// WMMA fp32 GEMM:  C[M,N] = A[M,K](row-major, lda) * W[N,K](row-major, ldw)^T
//                  (+ b1[N]) (+ b2[N]) (+ C if accumulate)
// One 16x16 C tile per wave. K consumed 32 at a time: 8x global_load_b64 per
// operand, then 8x v_wmma_f32_16x16x4_f32. Out-of-range lanes are clamped to
// row 0 (always valid memory); their garbage products land only in C entries
// the guarded epilogue never stores. K must be a multiple of 32
// (true for 4096 / 1024 / 512).
// ---------------------------------------------------------------------------
__global__ __launch_bounds__(256)
void wmma_gemm_nt(const float* __restrict__ A, int lda,
                  const float* __restrict__ W, int ldw,
                  float* __restrict__ C, int ldc,
                  int M, int N, int K,
                  const float* __restrict__ b1,
                  const float* __restrict__ b2,
                  int accumulate)
{
    const int wavesPerBlock = blockDim.x >> 5;
    const int waveId = blockIdx.x * wavesPerBlock + (threadIdx.x >> 5);
    const int lane   = threadIdx.x & 31;
    const int tilesN = (N + 15) >> 4;
    const int tilesM = (M + 15) >> 4;
    if (waveId >= tilesM * tilesN) return;   // wave-uniform: EXEC stays all-ones for WMMA
    const int tm = waveId / tilesN;
    const int tn = waveId % tilesN;

    const int rowA = tm * 16 + (lane & 15);          // A-frag: row in lane&15
    const int colW = tn * 16 + (lane & 15);          // W row == C column
    const int kHi  = (lane >> 4) * 2;                // K sub-pair selected by lane>>4
    const int cCol = tn * 16 + (lane & 15);          // C/D: col in lane&15
    const int cRowBase = tm * 16 + ((lane >> 4) ? 8 : 0);  // row = v + 8*(lane>>4)

    v8f acc = {};
    if (accumulate) {
#pragma unroll
        for (int v = 0; v < 8; ++v) {
            int r = cRowBase + v;
            acc[v] = (r < M && cCol < N) ? C[(size_t)r * ldc + cCol] : 0.0f;
        }
    }

    // Clamp to a valid row; no per-element predication in the hot loop.
    const float* __restrict__ pa =
        A + (size_t)(rowA < M ? rowA : 0) * lda + kHi;
    const float* __restrict__ pb =
        W + (size_t)(colW < N ? colW : 0) * ldw + kHi;

    for (int k0 = 0; k0 < K; k0 += 32) {
        v2f af[8], bf[8];
#pragma unroll
        for (int kk = 0; kk < 8; ++kk) {
            af[kk] = *(const v2f*)(pa + kk * 4);   // global_load_b64
            bf[kk] = *(const v2f*)(pb + kk * 4);   // global_load_b64
        }
#pragma unroll
        for (int kk = 0; kk < 8; ++kk) {
            acc = __builtin_amdgcn_wmma_f32_16x16x4_f32(
                /*neg_a=*/false, af[kk], /*neg_b=*/false, bf[kk],
                /*c_mod=*/(short)0, acc, /*reuse_a=*/false, /*reuse_b=*/false);
        }
        pa += 32;
        pb += 32;
    }

#pragma unroll
    for (int v = 0; v < 8; ++v) {
        int r = cRowBase + v;
        if (r < M && cCol < N) {
            float out = acc[v];
            if (b1) out += b1[cCol];
            if (b2) out += b2[cCol];
            C[(size_t)r * ldc + cCol] = out;
        }
    }
}

// ---------------------------------------------------------------------------
// Persistent single-workgroup LSTM recurrence.
// Gpre[t,2048] already contains x@Wih^T + bih + bhh. Each of 1024 threads owns
// two gate rows of Whh (GEMV from h in LDS); threads 0..511 own c[n]/h[n].
// ---------------------------------------------------------------------------
__global__ __launch_bounds__(1024)
void lstm_rec(const float* __restrict__ Gpre,
              const float* __restrict__ Whh,   // [2048, 512]
              const float* __restrict__ h0,    // [512]
              const float* __restrict__ c0,    // [512]
              float* __restrict__ outSeq,      // [T, 512]
              float* __restrict__ hT,          // [512]
              float* __restrict__ cT,          // [512]
              int T)
{
    __shared__ float h_lds[HID];
    __shared__ float gate_lds[GATES];
    const int tid = threadIdx.x;

    if (tid < HID) h_lds[tid] = h0[tid];
    float c = (tid < HID) ? c0[tid] : 0.0f;
    __syncthreads();

    for (int t = 0; t < T; ++t) {
#pragma unroll
        for (int gi = 0; gi < 2; ++gi) {
            const int g = tid + gi * 1024;
            const float* __restrict__ w = Whh + (size_t)g * HID;
            float acc = Gpre[(size_t)t * GATES + g];
#pragma unroll 8
            for (int j = 0; j < HID; j += 4) {
                float4 wv = *(const float4*)(w + j);
                acc += wv.x * h_lds[j]     + wv.y * h_lds[j + 1]
                     + wv.z * h_lds[j + 2] + wv.w * h_lds[j + 3];
            }
            gate_lds[g] = acc;
        }
        __syncthreads();
        if (tid < HID) {
            const float ig = gate_lds[tid];
            const float fg = gate_lds[HID + tid];
            const float gg = gate_lds[2 * HID + tid];
            const float og = gate_lds[3 * HID + tid];
            const float si = 1.0f / (1.0f + expf(-ig));
            const float sf = 1.0f / (1.0f + expf(-fg));
            const float so = 1.0f / (1.0f + expf(-og));
            c = sf * c + si * tanhf(gg);
            const float h = so * tanhf(c);
            h_lds[tid] = h;
            outSeq[(size_t)t * HID + tid] = h;
        }
        __syncthreads();
    }
    if (tid < HID) { hT[tid] = h_lds[tid]; cT[tid] = c; }
}

// Gates for an all-zero input: just bias broadcast.
__global__ void bias_fill(float* __restrict__ G,
                          const float* __restrict__ b1,
                          const float* __restrict__ b2, int total)
{
    int n = blockIdx.x * blockDim.x + threadIdx.x;
    if (n < total) {
        int g = n & (GATES - 1);
        G[n] = b1[g] + b2[g];
    }
}

// Teacher-forced one-hot embedding: emb[t] = W_emb[:, prev[t]] + b_emb
__global__ void embed_kernel(const int* __restrict__ target,
                             const float* __restrict__ W_emb,  // [512, 3003]
                             const float* __restrict__ b_emb,
                             float* __restrict__ E)
{
    const int t = blockIdx.x;      // 0..39
    const int r = threadIdx.x;     // 0..511
    const int idx = (t == 0) ? 0 : target[t - 1];
    E[(size_t)t * HID + r] = W_emb[(size_t)r * VOCAB + idx] + b_emb[r];
}

// In-place row-wise log_softmax.
__global__ __launch_bounds__(256)
void log_softmax_rows(float* __restrict__ X, int cols)
{
    __shared__ float red[256];
    float* __restrict__ x = X + (size_t)blockIdx.x * cols;
    const int tid = threadIdx.x;

    float m = -INFINITY;
    for (int j = tid; j < cols; j += 256) m = fmaxf(m, x[j]);
    red[tid] = m; __syncthreads();
    for (int s = 128; s > 0; s >>= 1) {
        if (tid < s) red[tid] = fmaxf(red[tid], red[tid + s]);
        __syncthreads();
    }
    m = red[0]; __syncthreads();

    float sum = 0.0f;
    for (int j = tid; j < cols; j += 256) sum += expf(x[j] - m);
    red[tid] = sum; __syncthreads();
    for (int s = 128; s > 0; s >>= 1) {
        if (tid < s) red[tid] += red[tid + s];
        __syncthreads();
    }
    const float lse = m + logf(red[0]);

    for (int j = tid; j < cols; j += 256) x[j] = x[j] - lse;
}

// ---------------------------------------------------------------------------
extern "C" void kernel_launch(void* const* d_in, const int* in_sizes, int n_in,
                              void* d_out, int out_size, void* d_ws, size_t ws_size,
                              hipStream_t stream)
{
    const float* input_feat = (const float*)d_in[0];   // [80,1,4096]
    const float* h1 = (const float*)d_in[1];           // [2,1,512]
    const float* c1 = (const float*)d_in[2];
    const float* h2 = (const float*)d_in[3];
    const float* c2 = (const float*)d_in[4];
    const float* l1_Wih0 = (const float*)d_in[5];      // [2048,4096]
    const float* l1_Whh0 = (const float*)d_in[6];      // [2048,512]
    const float* l1_bih0 = (const float*)d_in[7];
    const float* l1_bhh0 = (const float*)d_in[8];
    const float* l1_Wih1 = (const float*)d_in[9];      // [2048,512]
    const float* l1_Whh1 = (const float*)d_in[10];
    const float* l1_bih1 = (const float*)d_in[11];
    const float* l1_bhh1 = (const float*)d_in[12];
    const float* l2_Wih0 = (const float*)d_in[13];     // [2048,1024]
    const float* l2_Whh0 = (const float*)d_in[14];
    const float* l2_bih0 = (const float*)d_in[15];
    const float* l2_bhh0 = (const float*)d_in[16];
    const float* l2_Wih1 = (const float*)d_in[17];     // [2048,512]
    const float* l2_Whh1 = (const float*)d_in[18];
    const float* l2_bih1 = (const float*)d_in[19];
    const float* l2_bhh1 = (const float*)d_in[20];
    const float* W_out  = (const float*)d_in[21];      // [3003,512]
    const float* b_out  = (const float*)d_in[22];
    const float* W_emb  = (const float*)d_in[23];      // [512,3003]
    const float* b_emb  = (const float*)d_in[24];
    const int*   target = (const int*)d_in[25];        // [40]
    float* out = (float*)d_out;                        // [40,1,3003]

    // workspace layout (floats)
    float* ws = (float*)d_ws;
    float* G  = ws;                         // [80,2048]
    float* S0 = G  + (size_t)TIN * GATES;   // [80,512]
    float* S1 = S0 + (size_t)TIN * HID;     // [80,512]
    float* S2 = S1 + (size_t)TIN * HID;     // [80,512]
    float* st = S2 + (size_t)TIN * HID;     // 8 x [512] h/c states
    float* st_h0 = st + 0 * HID; float* st_c0 = st + 1 * HID;
    float* st_h1 = st + 2 * HID; float* st_c1 = st + 3 * HID;
    float* st_h2 = st + 4 * HID; float* st_c2 = st + 5 * HID;
    float* st_h3 = st + 6 * HID; float* st_c3 = st + 7 * HID;

    const float* NUL = nullptr;

    auto gemm = [&](const float* A, int lda, const float* W, int ldw,
                    float* C, int ldc, int M, int N, int K,
                    const float* b1, const float* b2, int acc) {
        int tiles  = ((M + 15) / 16) * ((N + 15) / 16);
        int blocks = (tiles + 7) / 8;   // 8 waves / 256-thread block
        wmma_gemm_nt<<<blocks, 256, 0, stream>>>(A, lda, W, ldw, C, ldc,
                                                 M, N, K, b1, b2, acc);
    };
    auto rec = [&](const float* Gp, const float* Whh, const float* h0,
                   const float* c0, float* o, float* hT, float* cT, int T) {
        lstm_rec<<<1, 1024, 0, stream>>>(Gp, Whh, h0, c0, o, hT, cT, T);
    };

    // ---------------- encode ----------------
    // lstm1 layer 0: K=4096 GEMM then recurrence
    gemm(input_feat, INDIM, l1_Wih0, INDIM, G, GATES, TIN, GATES, INDIM,
         l1_bih0, l1_bhh0, 0);
    rec(G, l1_Whh0, h1, c1, S0, st_h0, st_c0, TIN);
    // lstm1 layer 1
    gemm(S0, HID, l1_Wih1, HID, G, GATES, TIN, GATES, HID, l1_bih1, l1_bhh1, 0);
    rec(G, l1_Whh1, h1 + HID, c1 + HID, S1, st_h1, st_c1, TIN);
    // lstm2 layer 0: enc_in = [zeros(512); out1] -> only cols 512..1023 of Wih
    gemm(S1, HID, l2_Wih0 + HID, 2 * HID, G, GATES, TIN, GATES, HID,
         l2_bih0, l2_bhh0, 0);
    rec(G, l2_Whh0, h2, c2, S0, st_h2, st_c2, TIN);
    // lstm2 layer 1 (output discarded; states kept)
    gemm(S0, HID, l2_Wih1, HID, G, GATES, TIN, GATES, HID, l2_bih1, l2_bhh1, 0);
    rec(G, l2_Whh1, h2 + HID, c2 + HID, S1, st_h3, st_c3, TIN);

    // ---------------- decode ----------------
    // lstm1 layer 0 over zero input: gates = bias only
    {
        int total = TOUT * GATES;
        bias_fill<<<(total + 255) / 256, 256, 0, stream>>>(G, l1_bih0, l1_bhh0, total);
    }
    rec(G, l1_Whh0, st_h0, st_c0, S0, st_h0, st_c0, TOUT);
    // lstm1 layer 1
    gemm(S0, HID, l1_Wih1, HID, G, GATES, TOUT, GATES, HID, l1_bih1, l1_bhh1, 0);
    rec(G, l1_Whh1, st_h1, st_c1, S1, st_h1, st_c1, TOUT);   // S1 = out1d

    // embeddings (teacher forcing, one-hot column gather)
    embed_kernel<<<TOUT, HID, 0, stream>>>(target, W_emb, b_emb, S2);

    // lstm2 layer 0: dec_in = [emb ; out1d] -> split-K over Wih column halves
    gemm(S2, HID, l2_Wih0,       2 * HID, G, GATES, TOUT, GATES, HID,
         l2_bih0, l2_bhh0, 0);
    gemm(S1, HID, l2_Wih0 + HID, 2 * HID, G, GATES, TOUT, GATES, HID,
         NUL, NUL, 1);
    rec(G, l2_Whh0, st_h2, st_c2, S0, st_h2, st_c2, TOUT);
    // lstm2 layer 1
    gemm(S0, HID, l2_Wih1, HID, G, GATES, TOUT, GATES, HID, l2_bih1, l2_bhh1, 0);
    rec(G, l2_Whh1, st_h3, st_c3, S2, st_h3, st_c3, TOUT);   // S2 = out2d

    // logits straight into d_out, then in-place log_softmax
    gemm(S2, HID, W_out, HID, out, VOCAB, TOUT, VOCAB, HID, b_out, NUL, 0);
    log_softmax_rows<<<TOUT, 256, 0, stream>>>(out, VOCAB);
}